// MeshRasterizer_55465207661092
// MI455X (gfx1250) — compile-verified
//
#include <hip/hip_runtime.h>

#define HH 256
#define WW 256
#define NP (HH * WW)      // 65536 pixels
#define NV 4096
#define NF 4096
#define ND 32
#define EPSF 1e-8f

typedef __attribute__((ext_vector_type(2))) float v2f;
typedef __attribute__((ext_vector_type(8))) float v8f;

// ---------------------------------------------------------------- MVP = cam @ view
__global__ void k_mvp(const float* __restrict__ cam, const float* __restrict__ view,
                      float* __restrict__ mvp) {
    int t = threadIdx.x;
    if (t < 16) {
        int r = t >> 2, c = t & 3;
        float s = 0.0f;
        for (int k = 0; k < 4; ++k) s += cam[r * 4 + k] * view[k * 4 + c];
        mvp[t] = s;
    }
}

// ---------------------------------------------------------------- vertex transform
__global__ void k_vert(const float* __restrict__ verts, const float* __restrict__ mvp,
                       float* __restrict__ screen) {
    int v = blockIdx.x * blockDim.x + threadIdx.x;
    if (v >= NV) return;
    float x = verts[v * 3 + 0], y = verts[v * 3 + 1], z = verts[v * 3 + 2];
    float p[4];
#pragma unroll
    for (int i = 0; i < 4; ++i)
        p[i] = mvp[i * 4 + 0] * x + mvp[i * 4 + 1] * y + mvp[i * 4 + 2] * z + mvp[i * 4 + 3];
    float nx = p[0] / p[3];
    float ny = p[1] / p[3];
    float nz = p[2] / p[3];
    screen[v * 3 + 0] = (nx + 1.0f) * 0.5f * (float)WW;
    screen[v * 3 + 1] = (ny + 1.0f) * 0.5f * (float)HH;
    screen[v * 3 + 2] = nz;
}

// ---------------------------------------------------------------- per-face affine setup
// setup[f*16]: [Au,Bu,Cu,0, Av,Bv,Cv,0, Ad,Bd,Cd,0, 0,0,0,0]
__global__ void k_face(const int* __restrict__ faces, const float* __restrict__ screen,
                       float* __restrict__ setup) {
    int f = blockIdx.x * blockDim.x + threadIdx.x;
    if (f >= NF) return;
    int i0 = faces[f * 3 + 0], i1 = faces[f * 3 + 1], i2 = faces[f * 3 + 2];
    float v0x = screen[i0 * 3 + 0], v0y = screen[i0 * 3 + 1], z0 = screen[i0 * 3 + 2];
    float v1x = screen[i1 * 3 + 0], v1y = screen[i1 * 3 + 1], z1 = screen[i1 * 3 + 2];
    float v2x = screen[i2 * 3 + 0], v2y = screen[i2 * 3 + 1], z2 = screen[i2 * 3 + 2];

    float e1x = v1x - v0x, e1y = v1y - v0y;
    float e2x = v2x - v0x, e2y = v2y - v0y;
    bool front = (e1x * e2y - e1y * e2x) >= 0.0f;

    float dot00 = e2x * e2x + e2y * e2y;
    float dot01 = e2x * e1x + e2y * e1y;
    float dot11 = e1x * e1x + e1y * e1y;
    float inv_den = 1.0f / (dot00 * dot11 - dot01 * dot01 + EPSF);

    // u = (dot11*dot02 - dot01*dot12)*inv_den  is affine in pixel (x,y)
    float Au = (dot11 * e2x - dot01 * e1x) * inv_den;
    float Bu = (dot11 * e2y - dot01 * e1y) * inv_den;
    float Cu = -(Au * v0x + Bu * v0y);
    float Av = (dot00 * e1x - dot01 * e2x) * inv_den;
    float Bv = (dot00 * e1y - dot01 * e2y) * inv_den;
    float Cv = -(Av * v0x + Bv * v0y);
    // depth = z0 + u*(z1-z0) + v*(z2-z0)
    float dz1 = z1 - z0, dz2 = z2 - z0;
    float Ad = Au * dz1 + Av * dz2;
    float Bd = Bu * dz1 + Bv * dz2;
    float Cd = z0 + Cu * dz1 + Cv * dz2;

    if (!front) { Au = 0.0f; Bu = 0.0f; Cu = -3e38f; }   // u < 0 everywhere => never inside

    float* s = setup + (size_t)f * 16;
    s[0]  = Au; s[1]  = Bu; s[2]  = Cu; s[3]  = 0.0f;
    s[4]  = Av; s[5]  = Bv; s[6]  = Cv; s[7]  = 0.0f;
    s[8]  = Ad; s[9]  = Bd; s[10] = Cd; s[11] = 0.0f;
    s[12] = 0.0f; s[13] = 0.0f; s[14] = 0.0f; s[15] = 0.0f;
}

// ---------------------------------------------------------------- WMMA rasterizer
// One wave handles 16 consecutive pixels (one row segment), loops over 256 tiles
// of 16 faces; u/v/depth for the 16x16 (face x pixel) tile come from
// V_WMMA_F32_16X16X4_F32 with B = [x; y; 1; 0].
__global__ void k_raster(const float* __restrict__ setup,
                         float* __restrict__ bestD, int* __restrict__ bestF) {
    const int lane  = threadIdx.x & 31;
    const int wv    = (blockIdx.x * blockDim.x + threadIdx.x) >> 5;
    const int hlf   = lane >> 4;          // 0: K=0,1 rows | 1: K=2,3 rows
    const int li    = lane & 15;
    const int pbase = wv * 16;
    const int py    = pbase >> 8;         // / 256
    const int px    = (pbase & 255) + li;

    // B-matrix (4x16): rows [x; y; 1; 0], K pairs split across lane halves
    v2f B;
    B.x = hlf ? 1.0f : (float)px;
    B.y = hlf ? 0.0f : (float)py;

    float bd = __builtin_huge_valf();
    int   bf = -1;

    const v2f* su = (const v2f*)setup;    // 8 x v2f per face

    for (int t = 0; t < NF / 16; ++t) {
        const int face = t * 16 + li;                 // A-matrix row M = face-in-tile
        const long base = (long)face * 8 + hlf;       // half 0 -> (A,B), half 1 -> (C,0)
        v2f au = su[base + 0];
        v2f av = su[base + 2];
        v2f ad = su[base + 4];

        v8f zero = {};
        v8f Du = __builtin_amdgcn_wmma_f32_16x16x4_f32(false, au, false, B, (short)0, zero, false, false);
        v8f Dv = __builtin_amdgcn_wmma_f32_16x16x4_f32(false, av, false, B, (short)0, zero, false, false);
        v8f Dd = __builtin_amdgcn_wmma_f32_16x16x4_f32(false, ad, false, B, (short)0, zero, false, false);

        const int fb = t * 16 + hlf * 8;              // D VGPR m: M = m + 8*half
#pragma unroll
        for (int m = 0; m < 8; ++m) {
            float u = Du[m], v = Dv[m], d = Dd[m];
            // w = 1-u-v; w>=0 <=> s<=1, w<=1 <=> s>=0  (s = u+v)
            float s  = u + v;
            float mn = fminf(fminf(u, v), s);
            float mx = fmaxf(fmaxf(u, v), s);
            // inside && closer folded into one predicate; !inside or d==inf/NaN
            // never passes d < bd (strict <: first face wins on ties).
            bool upd = (mn >= 0.0f) && (mx <= 1.0f) && (d < bd);
            if (upd) { bd = d; bf = fb + m; }
        }
    }

    // merge the two lane halves (faces ==0..7 vs ==8..15 mod 16); lower fid wins ties
    float od = __shfl_xor(bd, 16, 32);
    int   of = __shfl_xor(bf, 16, 32);
    bool take = (od < bd) || ((od == bd) && ((unsigned)of < (unsigned)bf));
    if (take) { bd = od; bf = of; }

    if (hlf == 0) {
        int p = pbase + li;
        bestD[p] = bd; bestF[p] = bf;
    }
}

// ---------------------------------------------------------------- feature interpolation
// One wave per pixel; lane == feature index (ND == 32 == wave32).
// Barycentrics of the winning face are recomputed from the affine setup table.
__global__ void k_shade(const int* __restrict__ faces, const float* __restrict__ vfeat,
                        const float* __restrict__ setup,
                        const float* __restrict__ bestD, const int* __restrict__ bestF,
                        float* __restrict__ out) {
    const int tid  = blockIdx.x * blockDim.x + threadIdx.x;
    const int lane = tid & 31;
    const int p    = tid >> 5;
    if (p >= NP) return;

    int   fid  = bestF[p];
    float d    = bestD[p];
    bool  mask = fid >= 0;
    int   sf   = mask ? fid : 0;

    const float* s = setup + (size_t)sf * 16;
    float px = (float)(p & 255), py = (float)(p >> 8);
    float u = s[0] * px + s[1] * py + s[2];
    float v = s[4] * px + s[5] * py + s[6];
    float w = 1.0f - u - v;

    int i0 = faces[sf * 3 + 0], i1 = faces[sf * 3 + 1], i2 = faces[sf * 3 + 2];
    float feat = w * vfeat[i0 * ND + lane] + u * vfeat[i1 * ND + lane] + v * vfeat[i2 * ND + lane];
    feat = mask ? feat : 0.0f;

    out[(long)p * 33 + lane] = feat;
    if (lane == 0) {
        out[(long)p * 33 + 32] = mask ? d : 0.0f;            // depth_out
        out[(long)NP * 33 + p] = (float)fid;                  // fbuf
        out[(long)NP * 34 + p] = mask ? 1.0f : 0.0f;          // mask
    }
}

// ---------------------------------------------------------------- launch
extern "C" void kernel_launch(void* const* d_in, const int* in_sizes, int n_in,
                              void* d_out, int out_size, void* d_ws, size_t ws_size,
                              hipStream_t stream) {
    const float* verts = (const float*)d_in[0];  // (4096,3) f32
    const int*   faces = (const int*)d_in[1];    // (4096,3) i32
    const float* vfeat = (const float*)d_in[2];  // (4096,32) f32
    const float* cam   = (const float*)d_in[3];  // (4,4) f32
    const float* view  = (const float*)d_in[4];  // (4,4) f32
    float* out = (float*)d_out;

    char* ws = (char*)d_ws;
    size_t o = 0;
    float* mvp    = (float*)(ws + o); o += 256;                 // 16 f32 (padded)
    float* screen = (float*)(ws + o); o += (size_t)NV * 3 * 4;  // 48 KB
    float* setup  = (float*)(ws + o); o += (size_t)NF * 16 * 4; // 256 KB
    float* bestD  = (float*)(ws + o); o += (size_t)NP * 4;
    int*   bestF  = (int*)  (ws + o); o += (size_t)NP * 4;
    (void)ws_size; (void)in_sizes; (void)n_in; (void)out_size;

    k_mvp   <<<1, 32, 0, stream>>>(cam, view, mvp);
    k_vert  <<<NV / 256, 256, 0, stream>>>(verts, mvp, screen);
    k_face  <<<NF / 256, 256, 0, stream>>>(faces, screen, setup);
    k_raster<<<(NP / 16) / 8, 256, 0, stream>>>(setup, bestD, bestF);
    k_shade <<<(NP * 32) / 256, 256, 0, stream>>>(faces, vfeat, setup, bestD, bestF, out);
}